// DotGATLayer_54932631716064
// MI455X (gfx1250) — compile-verified
//
#include <hip/hip_runtime.h>
#include <math.h>

typedef __attribute__((ext_vector_type(2))) float v2f;
typedef __attribute__((ext_vector_type(8))) float v8f;

#define DIM       128
#define N_NODES   50000
#define N_EDGES   800000
#define TAU       0.08838834764831845f   /* 1/sqrt(128) */
#define M_TILES   (N_NODES / 16)         /* 3125 */
#define BLOCKS_X  125                    /* persistent: 25 m-tiles per block */
#define TILES_PER_BLOCK (M_TILES / BLOCKS_X)

// ---------------------------------------------------------------------------
// Phase 1: q,k,v = z @ W{q,k,v} + b  using V_WMMA_F32_16X16X4_F32.
// One wave owns a 16-column N tile and holds all 32 B fragments (64 VGPRs)
// in registers, then loops over 25 m-tiles; steady-state inner loop is
// 1 x global_load_b64 (A frag) per WMMA.
// ---------------------------------------------------------------------------
__global__ __launch_bounds__(256) void proj_kernel(
    const float* __restrict__ z,
    const float* __restrict__ Wq, const float* __restrict__ bq,
    const float* __restrict__ Wk, const float* __restrict__ bk,
    const float* __restrict__ Wv, const float* __restrict__ bv,
    float* __restrict__ q, float* __restrict__ k, float* __restrict__ v)
{
    const int lane = threadIdx.x & 31;
    const int wave = threadIdx.x >> 5;          // 0..7 -> N tile
    const int mat  = blockIdx.y;                // 0=q 1=k 2=v

    const float* W; const float* b; float* out;
    if      (mat == 0) { W = Wq; b = bq; out = q; }
    else if (mat == 1) { W = Wk; b = bk; out = k; }
    else               { W = Wv; b = bv; out = v; }

    const int n0    = wave * 16;
    const int l15   = lane & 15;
    const int hi    = lane >> 4;                // 0: K=k0,k0+1 ; 1: K=k0+2,k0+3
    const int ncol  = n0 + l15;                 // B/C/D column for this lane
    const int khalf = hi * 2;

    // Load all B fragments once (W row-major, W[k][n]); kept in VGPRs.
    v2f Bf[DIM / 4];
#pragma unroll
    for (int t = 0; t < DIM / 4; ++t) {
        const int kk = t * 4;
        Bf[t].x = W[(size_t)(kk + khalf    ) * DIM + ncol];
        Bf[t].y = W[(size_t)(kk + khalf + 1) * DIM + ncol];
    }
    const float bias = b[ncol];

    for (int tile = 0; tile < TILES_PER_BLOCK; ++tile) {
        const int m0   = (blockIdx.x * TILES_PER_BLOCK + tile) * 16;
        const int mrow = m0 + l15;              // A-fragment row for this lane
        const float* zrow = z + (size_t)mrow * DIM + khalf;

        v8f c = {0.f, 0.f, 0.f, 0.f, 0.f, 0.f, 0.f, 0.f};
#pragma unroll
        for (int t = 0; t < DIM / 4; ++t) {
            // A 16x4 fragment (z row-major): lane holds 2 consecutive K values
            v2f a; a.x = zrow[t * 4]; a.y = zrow[t * 4 + 1];
            // D = A x B + C  (8 args: neg_a, A, neg_b, B, c_mod, C, reuse_a, reuse_b)
            c = __builtin_amdgcn_wmma_f32_16x16x4_f32(false, a, false, Bf[t],
                                                      (short)0, c, false, false);
        }

#pragma unroll
        for (int j = 0; j < 8; ++j) {
            const int row = m0 + j + (hi << 3); // C/D layout: M = j (+8 for hi lanes)
            out[(size_t)row * DIM + ncol] = c[j] + bias;
        }
    }
}

// ---------------------------------------------------------------------------
// Init: out = 0, denom = 0, m = -inf
// ---------------------------------------------------------------------------
__global__ void init_kernel(float* __restrict__ out, float* __restrict__ m,
                            float* __restrict__ denom)
{
    const int i = blockIdx.x * blockDim.x + threadIdx.x;
    if (i < N_NODES * DIM) out[i] = 0.0f;
    if (i < N_NODES) { m[i] = -INFINITY; denom[i] = 0.0f; }
}

// Monotonic float atomic-max via signed/unsigned integer ordering trick.
__device__ inline void atomicMaxFloat(float* addr, float val)
{
    if (val >= 0.0f) atomicMax((int*)addr, __float_as_int(val));
    else             atomicMin((unsigned int*)addr, __float_as_uint(val));
}

// ---------------------------------------------------------------------------
// Phase 2: e[i] = tau * dot(k[src], q[dst]); segment-max into m[dst].
// One wave per edge; 4 floats/lane; gathers are L2-resident (q+k+v = 77MB < 192MB).
// ---------------------------------------------------------------------------
__global__ __launch_bounds__(256) void edge_logit_kernel(
    const float* __restrict__ q, const float* __restrict__ k,
    const int* __restrict__ src, const int* __restrict__ dst,
    float* __restrict__ e, float* __restrict__ m)
{
    const int lane = threadIdx.x & 31;
    const int edge = blockIdx.x * 8 + (threadIdx.x >> 5);
    if (edge >= N_EDGES) return;

    const int s = src[edge];
    const int d = dst[edge];

    const float4 kv = ((const float4*)(k + (size_t)s * DIM))[lane];
    const float4 qv = ((const float4*)(q + (size_t)d * DIM))[lane];
    float acc = kv.x*qv.x + kv.y*qv.y + kv.z*qv.z + kv.w*qv.w;

#pragma unroll
    for (int off = 16; off > 0; off >>= 1)
        acc += __shfl_xor(acc, off, 32);

    if (lane == 0) {
        const float val = acc * TAU;
        e[edge] = val;
        atomicMaxFloat(m + d, val);
    }
}

// ---------------------------------------------------------------------------
// Phase 3: ex = exp(e - m[dst]); denom[dst] += ex; out[dst] += ex * v[src]
// ---------------------------------------------------------------------------
__global__ __launch_bounds__(256) void edge_agg_kernel(
    const float* __restrict__ v,
    const int* __restrict__ src, const int* __restrict__ dst,
    const float* __restrict__ e, const float* __restrict__ m,
    float* __restrict__ denom, float* __restrict__ out)
{
    const int lane = threadIdx.x & 31;
    const int edge = blockIdx.x * 8 + (threadIdx.x >> 5);
    if (edge >= N_EDGES) return;

    const int s = src[edge];
    const int d = dst[edge];

    float mi = m[d];
    if (isinf(mi)) mi = 0.0f;                 // isolated-node guard (matches ref)
    const float w = __expf(e[edge] - mi);

    if (lane == 0) atomicAdd(denom + d, w);

    const float4 vv = ((const float4*)(v + (size_t)s * DIM))[lane];
    float* op = out + (size_t)d * DIM + lane * 4;
    atomicAdd(op + 0, w * vv.x);
    atomicAdd(op + 1, w * vv.y);
    atomicAdd(op + 2, w * vv.z);
    atomicAdd(op + 3, w * vv.w);
}

// ---------------------------------------------------------------------------
// Phase 4: out[n,:] /= max(denom[n], ->1 if zero)
// ---------------------------------------------------------------------------
__global__ void finalize_kernel(float* __restrict__ out,
                                const float* __restrict__ denom)
{
    const int i = blockIdx.x * blockDim.x + threadIdx.x;
    if (i >= N_NODES * DIM) return;
    const float dn = denom[i / DIM];
    out[i] *= (dn > 0.0f) ? (1.0f / dn) : 1.0f;
}

// ---------------------------------------------------------------------------
extern "C" void kernel_launch(void* const* d_in, const int* in_sizes, int n_in,
                              void* d_out, int out_size, void* d_ws, size_t ws_size,
                              hipStream_t stream)
{
    const float* z  = (const float*)d_in[0];
    const float* Wq = (const float*)d_in[1];
    const float* bq = (const float*)d_in[2];
    const float* Wk = (const float*)d_in[3];
    const float* bk = (const float*)d_in[4];
    const float* Wv = (const float*)d_in[5];
    const float* bv = (const float*)d_in[6];
    const int*  src = (const int*)d_in[7];
    const int*  dst = (const int*)d_in[8];
    float* out = (float*)d_out;

    // Workspace layout (floats): q | k | v | e | m | denom  (~80.5 MB)
    float* ws = (float*)d_ws;
    float* q     = ws;
    float* k     = q + (size_t)N_NODES * DIM;
    float* v     = k + (size_t)N_NODES * DIM;
    float* e     = v + (size_t)N_NODES * DIM;
    float* m     = e + (size_t)N_EDGES;
    float* denom = m + (size_t)N_NODES;

    // 1) projections via WMMA: persistent blocks, weights held in registers
    proj_kernel<<<dim3(BLOCKS_X, 3), 256, 0, stream>>>(
        z, Wq, bq, Wk, bk, Wv, bv, q, k, v);

    // 2) init accumulators / output
    init_kernel<<<(N_NODES * DIM + 255) / 256, 256, 0, stream>>>(out, m, denom);

    // 3) edge logits + segment max  (8 edges per 256-thread block)
    edge_logit_kernel<<<(N_EDGES + 7) / 8, 256, 0, stream>>>(q, k, src, dst, e, m);

    // 4) softmax numerator/denominator aggregation
    edge_agg_kernel<<<(N_EDGES + 7) / 8, 256, 0, stream>>>(v, src, dst, e, m, denom, out);

    // 5) normalize
    finalize_kernel<<<(N_NODES * DIM + 255) / 256, 256, 0, stream>>>(out, denom);
}